// Grouper_10591389352196
// MI455X (gfx1250) — compile-verified
//
#include <hip/hip_runtime.h>
#include <stdint.h>

typedef __attribute__((ext_vector_type(2))) float v2f;
typedef __attribute__((ext_vector_type(8))) float v8f;

#define B_ 16
#define N_ 8192
#define G_ 512
#define K_ 32
#define BIGF 3.0e37f

// ---------------------------------------------------------------------------
// Kernel 1: farthest point sampling, one workgroup per batch.
// The whole batch point cloud (96KB) is staged into LDS once via
// global_load_async_to_lds_b128; all 512 sequential rounds then run at LDS
// latency. dists[] also lives in LDS.
// ---------------------------------------------------------------------------
__global__ __launch_bounds__(256) void fps_kernel(const float* __restrict__ xyz,
                                                  float* __restrict__ out_center,
                                                  float* __restrict__ centers4) {
  __shared__ __align__(16) float pts[N_ * 3];   // 96 KB, AoS x,y,z
  __shared__ float dists[N_];                   // 32 KB
  __shared__ float rv[256];
  __shared__ int   ri[256];
  __shared__ int   cur;

  const int b   = blockIdx.x;
  const int tid = threadIdx.x;
  const float* base = xyz + (size_t)b * N_ * 3;

  // ---- async bulk copy: 96KB global -> LDS (256 lanes x 24 x b128) ----
  {
    const unsigned lds0 = (unsigned)(uintptr_t)pts + (unsigned)tid * 16u;
    const uint64_t ga0  = (uint64_t)(uintptr_t)base + (uint64_t)tid * 16u;
#pragma unroll 4
    for (int it = 0; it < (N_ * 3 * 4) / (256 * 16); ++it) {
      unsigned lds = lds0 + (unsigned)it * 4096u;
      uint64_t ga  = ga0 + (uint64_t)it * 4096u;
      asm volatile("global_load_async_to_lds_b128 %0, %1, off"
                   :: "v"(lds), "v"(ga) : "memory");
    }
    asm volatile("s_wait_asynccnt 0" ::: "memory");
  }

  if (tid == 0) cur = 0;
  for (int n = tid; n < N_; n += 256) dists[n] = 1e10f;
  __syncthreads();

  for (int g = 0; g < G_; ++g) {
    const int f = cur;
    const float cx = pts[f * 3 + 0];
    const float cy = pts[f * 3 + 1];
    const float cz = pts[f * 3 + 2];
    if (tid == 0) {
      const size_t o = (size_t)b * G_ + g;
      out_center[o * 3 + 0] = cx;
      out_center[o * 3 + 1] = cy;
      out_center[o * 3 + 2] = cz;
      centers4[o * 4 + 0] = cx;
      centers4[o * 4 + 1] = cy;
      centers4[o * 4 + 2] = cz;
      centers4[o * 4 + 3] = 0.0f;
    }

    float bv = -1.0f;
    int   bi = 0;
    for (int n = tid; n < N_; n += 256) {
      const float dx = pts[n * 3 + 0] - cx;
      const float dy = pts[n * 3 + 1] - cy;
      const float dz = pts[n * 3 + 2] - cz;
      const float d  = dx * dx + dy * dy + dz * dz;
      const float dm = fminf(dists[n], d);
      dists[n] = dm;
      if (dm > bv) { bv = dm; bi = n; }
    }
    rv[tid] = bv;
    ri[tid] = bi;
    __syncthreads();
    for (int s = 128; s > 0; s >>= 1) {
      if (tid < s) {
        if (rv[tid + s] > rv[tid]) { rv[tid] = rv[tid + s]; ri[tid] = ri[tid + s]; }
      }
      __syncthreads();
    }
    if (tid == 0) cur = ri[0];
    __syncthreads();
  }
}

// ---------------------------------------------------------------------------
// Kernel 2: KNN (top-32 by squared distance) via V_WMMA_F32_16X16X4_F32.
// Per 128-point chunk each wave async-stages its own 16 points into LDS
// (wave-local, so only s_wait_asynccnt is needed, no block barrier), builds
// the 4x16 B operand branch-free, WMMAs the -2*c.p cross-term tile, and the
// 256 selector threads (16 centers x 16 sub-selectors) keep register-resident
// top-32 lists. Final 32-round LDS min-merge emits re-centered neighborhoods.
// ---------------------------------------------------------------------------
__global__ __launch_bounds__(256) void knn_kernel(const float* __restrict__ xyz,
                                                  const float* __restrict__ centers4,
                                                  float* __restrict__ out_nb) {
  __shared__ float sq[16][129];                  // padded: kill 16-way bank conflicts
  __shared__ __align__(16) float stage[8][48];   // 16 points x xyz per wave
  __shared__ float csx[16], csy[16], csz[16], cn[16];
  __shared__ float mval[256];
  __shared__ int   mpidx[256];

  const int tid  = threadIdx.x;
  const int b    = blockIdx.x >> 5;
  const int g0   = (blockIdx.x & 31) << 4;
  const int lane = tid & 31;
  const int w    = tid >> 5;  // wave id 0..7
  const float* base = xyz + (size_t)b * N_ * 3;

  if (tid < 16) {
    const size_t o = ((size_t)b * G_ + g0 + tid) * 4;
    const float x = centers4[o + 0], y = centers4[o + 1], z = centers4[o + 2];
    csx[tid] = x; csy[tid] = y; csz[tid] = z;
    cn[tid] = x * x + y * y + z * z;
  }
  __syncthreads();

  // A matrix (16x4 fp32): lanes 0-15 carry K=0,1 (x,y); lanes 16-31 carry K=2,3 (z,0)
  const int ma = lane & 15;
  v2f a;
  a.x = (lane < 16) ? csx[ma] : csz[ma];
  a.y = (lane < 16) ? csy[ma] : 0.0f;

  v8f zeroC = {};

  float vals[K_];
  int   idxs[K_];
#pragma unroll
  for (int i = 0; i < K_; ++i) { vals[i] = BIGF; idxs[i] = 0; }
  float curmax  = BIGF;
  int   maxslot = 0;

  const int msel = tid & 15;   // which center this selector owns
  const int ssel = tid >> 4;   // sub-selector 0..15

  const unsigned stage_lds = (unsigned)(uintptr_t)&stage[w][0];

  for (int chunk = 0; chunk < N_ / 128; ++chunk) {
    // ---- async-stage this wave's 16 points (192B, 16B-aligned) ----
    if (lane < 12) {
      unsigned lds = stage_lds + (unsigned)lane * 16u;
      uint64_t ga  = (uint64_t)(uintptr_t)base +
                     (uint64_t)(chunk * 128 + w * 16) * 12u + (uint64_t)lane * 16u;
      asm volatile("global_load_async_to_lds_b128 %0, %1, off"
                   :: "v"(lds), "v"(ga) : "memory");
    }
    asm volatile("s_wait_asynccnt 0" ::: "memory");

    // ---- B matrix (4x16 fp32), branch-free from staged LDS ----
    const int i16 = lane & 15;
    const float px = stage[w][3 * i16 + 0];
    const float py = stage[w][3 * i16 + 1];
    const float pz = stage[w][3 * i16 + 2];
    const float pn = px * px + py * py + pz * pz;

    v2f bb;
    bb.x = (lane < 16) ? px : pz;
    bb.y = (lane < 16) ? py : 0.0f;

    // 16x16 tile of dot(c_m, p_n)
    v8f acc = __builtin_amdgcn_wmma_f32_16x16x4_f32(
        false, a, false, bb, (short)0, zeroC, false, false);

    const int mbase = (lane < 16) ? 0 : 8;
    const int col   = w * 16 + i16;
#pragma unroll
    for (int r = 0; r < 8; ++r) {
      sq[mbase + r][col] = cn[mbase + r] + pn - 2.0f * acc[r];
    }
    __syncthreads();

    // ---- selection: each thread scans 8 columns of its center's row ----
    const int jb = ssel * 8;
#pragma unroll
    for (int jj = 0; jj < 8; ++jj) {
      const float d   = sq[msel][jb + jj];
      const int   idx = chunk * 128 + jb + jj;
      if (d < curmax) {
#pragma unroll
        for (int i = 0; i < K_; ++i) {
          const bool hit = (i == maxslot);
          vals[i] = hit ? d : vals[i];
          idxs[i] = hit ? idx : idxs[i];
        }
        curmax = -BIGF;
#pragma unroll
        for (int i = 0; i < K_; ++i) {
          if (vals[i] > curmax) { curmax = vals[i]; maxslot = i; }
        }
      }
    }
    __syncthreads();
  }

  // ---- merge 16 partial top-32 lists per center, emit in ascending order ----
  const float cxm = csx[msel], cym = csy[msel], czm = csz[msel];
  for (int k = 0; k < K_; ++k) {
    float lmin = BIGF; int lidx = 0, lslot = 0;
#pragma unroll
    for (int i = 0; i < K_; ++i) {
      if (vals[i] < lmin) { lmin = vals[i]; lidx = idxs[i]; lslot = i; }
    }
    mval[msel * 16 + ssel]  = lmin;
    mpidx[msel * 16 + ssel] = lidx;
    __syncthreads();

    float bmin = BIGF * 1.1f; int bs = 0;
    for (int ss = 0; ss < 16; ++ss) {
      const float v = mval[msel * 16 + ss];
      if (v < bmin) { bmin = v; bs = ss; }
    }
    const int wp = mpidx[msel * 16 + bs];
    if (ssel == 0) {
      const float* pp = base + (size_t)wp * 3;
      const size_t o = (((size_t)b * G_ + g0 + msel) * K_ + k) * 3;
      out_nb[o + 0] = pp[0] - cxm;
      out_nb[o + 1] = pp[1] - cym;
      out_nb[o + 2] = pp[2] - czm;
    }
    if (ssel == bs) {
#pragma unroll
      for (int i = 0; i < K_; ++i) vals[i] = (i == lslot) ? BIGF : vals[i];
    }
    __syncthreads();
  }
}

// ---------------------------------------------------------------------------
extern "C" void kernel_launch(void* const* d_in, const int* in_sizes, int n_in,
                              void* d_out, int out_size, void* d_ws, size_t ws_size,
                              hipStream_t stream) {
  (void)in_sizes; (void)n_in; (void)out_size; (void)ws_size;
  const float* xyz = (const float*)d_in[0];
  float* nb  = (float*)d_out;                                   // [B,G,K,3]
  float* ctr = (float*)d_out + (size_t)B_ * G_ * K_ * 3;        // [B,G,3]
  float* c4  = (float*)d_ws;                                    // [B,G,4] padded centers

  fps_kernel<<<B_, 256, 0, stream>>>(xyz, ctr, c4);
  knn_kernel<<<B_ * (G_ / 16), 256, 0, stream>>>(xyz, c4, nb);
}